// RoiAlignPyTorch_14929306321673
// MI455X (gfx1250) — compile-verified
//
#include <hip/hip_runtime.h>

// ROI-Align for MI455X (gfx1250), separable-interpolation formulation:
//   out[c] = Wy (7x16 pad 16x16) * patch[c] (16x16) * Wx^T (16x7)
// via V_WMMA_F32_16X16X4_F32 chains (8 wmma / channel), with the 16x16x64
// feature patch DMA'd into LDS by the Tensor Data Mover (tensor_load_to_lds,
// D# built per CDNA5 ISA). TDM OOB-zero-fill provides edge padding.

typedef __attribute__((ext_vector_type(2))) float v2f;
typedef __attribute__((ext_vector_type(8))) float v8f;
typedef __attribute__((ext_vector_type(4))) unsigned int v4u;
typedef __attribute__((ext_vector_type(8))) unsigned int v8u;

#define PH 7
#define PW 7
#define SMAX 2
#define SCALEF 0.0625f
#define CCH 256
#define HF 152
#define WF 152
#define CH_CHUNK 64
#define NWAVE 8
#define USTRIDE 17   // padded stride for U transpose tile: conflict-free reads

// dynamic-LDS layout (float offsets): sP at 0 so D#.lds_addr == 0 exactly
#define SP_FLOATS   (CH_CHUNK * 16 * 16)          // 16384
#define SWY_OFF     (SP_FLOATS)                   // 16384
#define SWX_OFF     (SWY_OFF + 256)               // 16640
#define SU_OFF      (SWX_OFF + 256)               // 16896
#define LDS_FLOATS  (SU_OFF + NWAVE * 16 * USTRIDE)
#define LDS_BYTES   (LDS_FLOATS * 4)              // 76288 B

__global__ __launch_bounds__(256) void roi_align_wmma(
    const float* __restrict__ feat, const float* __restrict__ rois,
    float* __restrict__ out, int K)
{
    extern __shared__ float smem[];
    float* sP  = smem;               // 64 ch x 16 x 16 patch (offset 0)
    float* sWy = smem + SWY_OFF;     // y interp+avg matrix (rows 7..15 = 0)
    float* sWx = smem + SWX_OFF;     // x interp+avg matrix
    float* sU  = smem + SU_OFF;      // per-wave U transpose staging

    const int tid  = threadIdx.x;
    const int lane = tid & 31;
    const int wv   = tid >> 5;
    const int half = (lane >> 4) & 1;
    const int l16  = lane & 15;
    const int k    = blockIdx.x;        // roi index
    const int c0   = blockIdx.y * CH_CHUNK;
    if (k >= K) return;                 // uniform

    // ---- ROI parameters (redundant per thread, deterministic) ----
    const float r0  = rois[k * 5 + 0];
    const float fx1 = rois[k * 5 + 1] * SCALEF;
    const float fy1 = rois[k * 5 + 2] * SCALEF;
    const float fx2 = rois[k * 5 + 3] * SCALEF;
    const float fy2 = rois[k * 5 + 4] * SCALEF;
    const int   b   = (int)r0;
    const float roi_w = fmaxf(fx2 - fx1, 1.0f);
    const float roi_h = fmaxf(fy2 - fy1, 1.0f);
    const float bin_h = roi_h / (float)PH;
    const float bin_w = roi_w / (float)PW;
    const float gh = fminf(fmaxf(ceilf(bin_h), 1.0f), (float)SMAX);
    const float gw = fminf(fmaxf(ceilf(bin_w), 1.0f), (float)SMAX);

    // Patch origin = floor index of the first (minimum) sample coordinate.
    // All tap indices fall in [origin, origin+15] (span <= roi + 2 <= 16).
    auto origin = [](float start, float bin, float g, int size) -> int {
        float c = fmaxf(start + 0.5f * bin / g, 0.0f);
        float low0 = fminf(floorf(c), (float)(size - 1));
        return (int)low0;
    };
    const int y0 = origin(fy1, bin_h, gh, HF);
    const int x0 = origin(fx1, bin_w, gw, WF);

    // ---- TDM: DMA the 16(x) x 16(y) x 64(ch) f32 tile into LDS starting at byte 0 ----
    // D# per CDNA5 ISA: group0 = {count, lds_addr, global_addr, type=2},
    // group1 = {data_size=4B, tensor_dim0/1 (clipped -> OOB zero pad),
    //           tile 16x16x64, stride0=W, stride1=H*W}, group2 = {tensor_dim2}.
    if (tid == 0) {
        const unsigned long long gaddr =
            (unsigned long long)(uintptr_t)feat +
            4ull * ((unsigned long long)(b * CCH + c0) * (HF * WF) +
                    (unsigned long long)y0 * WF + (unsigned long long)x0);
        const unsigned glo   = __builtin_amdgcn_readfirstlane((unsigned)gaddr);
        const unsigned ghi   = __builtin_amdgcn_readfirstlane((unsigned)(gaddr >> 32));
        const unsigned tdim0 = __builtin_amdgcn_readfirstlane((unsigned)(WF - x0));
        const unsigned tdim1 = __builtin_amdgcn_readfirstlane((unsigned)(HF - y0));

        v4u g0;
        g0[0] = 1u;                                   // count=1 (valid descriptor)
        g0[1] = 0u;                                   // lds_addr = 0 (sP)
        g0[2] = glo;                                  // global_addr[31:0]
        g0[3] = (ghi & 0x01FFFFFFu) | (2u << 30);     // global_addr[56:32] | type=2

        v8u g1;
        g1[0] = 0x00020000u;                          // data_size=2 (4B), mask=0
        g1[1] = (tdim0 & 0xFFFFu) << 16;              // tensor_dim0[15:0]
        g1[2] = ((tdim0 >> 16) & 0xFFFFu)             // tensor_dim0[31:16]
              | ((tdim1 & 0xFFFFu) << 16);            // tensor_dim1[15:0]
        g1[3] = ((tdim1 >> 16) & 0xFFFFu)             // tensor_dim1[31:16]
              | (16u << 16);                          // tile_dim0 = 16
        g1[4] = 16u | ((unsigned)CH_CHUNK << 16);     // tile_dim1=16, tile_dim2=64
        g1[5] = (unsigned)WF;                         // tensor_dim0_stride lo32
        g1[6] = (((unsigned)(HF * WF)) & 0xFFFFu) << 16; // stride0 hi16=0 | stride1[15:0]
        g1[7] = ((unsigned)(HF * WF)) >> 16;          // tensor_dim1_stride[47:16]

        v4u g2;
        g2[0] = (unsigned)CH_CHUNK;                   // tensor_dim2 = 64
        g2[1] = 0u; g2[2] = 0u; g2[3] = 0u;           // no dim3 / no iterate

        v4u g3;
        g3[0] = 0u; g3[1] = 0u; g3[2] = 0u; g3[3] = 0u;

        asm volatile("tensor_load_to_lds %0, %1, %2, %3"
                     :: "s"(g0), "s"(g1), "s"(g2), "s"(g3)
                     : "memory");
    }

    // ---- Build Wy / Wx while the DMA runs (wave 0: halves -> y/x axis) ----
    if (wv == 0) {
        const int   row   = l16;
        const int   axisx = half;
        const float start = axisx ? fx1 : fy1;
        const float bin   = axisx ? bin_w : bin_h;
        const float g     = axisx ? gw : gh;
        const int   size  = axisx ? WF : HF;
        const int   org   = axisx ? x0 : y0;
        float* Wrow = (axisx ? sWx : sWy) + row * 16;
        #pragma unroll
        for (int d = 0; d < 16; ++d) Wrow[d] = 0.0f;
        if (row < PH) {
            #pragma unroll
            for (int i = 0; i < SMAX; ++i) {
                const float smask = ((float)i < g) ? 1.0f : 0.0f;
                const float coord = start + (float)row * bin + ((float)i + 0.5f) * bin / g;
                const float valid = (coord > -1.0f && coord < (float)size) ? 1.0f : 0.0f;
                const float c     = fmaxf(coord, 0.0f);
                const float low0  = floorf(c);
                const int   lo    = (int)fminf(low0, (float)(size - 1));
                const int   hi    = (int)fminf(low0 + 1.0f, (float)(size - 1));
                const float frac  = (low0 >= (float)(size - 1)) ? 0.0f : (c - low0);
                const float wbase = smask * valid / g;   // fold 1/g averaging per axis
                const int dlo = min(max(lo - org, 0), 15);
                const int dhi = min(max(hi - org, 0), 15);
                Wrow[dlo] += (1.0f - frac) * wbase;
                Wrow[dhi] += frac * wbase;
            }
        }
    }

    // Drain the TDM transfer (no-op for waves that issued nothing), then sync.
    __builtin_amdgcn_s_wait_tensorcnt(0);
    __syncthreads();

    // ---- A-fragments (ISA 7.12.2: 16x4 A: lane=M, VGPR0=K{0,2}, VGPR1=K{1,3}) ----
    v2f aY[4], aX[4];
    #pragma unroll
    for (int j = 0; j < 4; ++j) {
        const int kk = 4 * j + 2 * half;
        aY[j].x = sWy[l16 * 16 + kk];
        aY[j].y = sWy[l16 * 16 + kk + 1];
        aX[j].x = sWx[l16 * 16 + kk];
        aX[j].y = sWx[l16 * 16 + kk + 1];
    }

    // ---- Per-wave channel loop: 8 channels each, 8 wmma per channel ----
    float* Uw = sU + wv * 16 * USTRIDE;
    #pragma unroll 1
    for (int cc = 0; cc < CH_CHUNK / NWAVE; ++cc) {
        const int cl = wv * (CH_CHUNK / NWAVE) + cc;
        const float* P = sP + cl * 256;

        // Stage A: U = Wy x P   (B 4x16: VGPR0=K{0,2} rows, VGPR1=K{1,3})
        v8f u = {};
        #pragma unroll
        for (int j = 0; j < 4; ++j) {
            const int kk = 4 * j + 2 * half;
            v2f bf;
            bf.x = P[kk * 16 + l16];
            bf.y = P[(kk + 1) * 16 + l16];
            u = __builtin_amdgcn_wmma_f32_16x16x4_f32(
                    false, aY[j], false, bf, (short)0, u, false, false);
        }

        // Transpose U through LDS (C/D layout: u[i] = U[i + 8*half][l16])
        #pragma unroll
        for (int i = 0; i < 8; ++i)
            Uw[(i + 8 * half) * USTRIDE + l16] = u[i];
        asm volatile("s_wait_dscnt 0" ::: "memory");   // intra-wave LDS RAW

        // Stage B: D = Wx x U^T  -> D[M=px][N=py]
        v8f d = {};
        #pragma unroll
        for (int j = 0; j < 4; ++j) {
            const int kk = 4 * j + 2 * half;
            v2f bf;
            bf.x = Uw[l16 * USTRIDE + kk];       // U^T[kk][l16] = U[l16][kk]
            bf.y = Uw[l16 * USTRIDE + kk + 1];
            d = __builtin_amdgcn_wmma_f32_16x16x4_f32(
                    false, aX[j], false, bf, (short)0, d, false, false);
        }

        // Write: lanes 0..6 hold py=l16, d[i] = out[py][px=i] for i<7 (half==0)
        if (half == 0 && l16 < PH) {
            float* op = out + ((size_t)k * CCH + (c0 + cl)) * (PH * PW) + l16 * PW;
            #pragma unroll
            for (int i = 0; i < PW; ++i)
                op[i] = d[i];
        }
    }
}

extern "C" void kernel_launch(void* const* d_in, const int* in_sizes, int n_in,
                              void* d_out, int out_size, void* d_ws, size_t ws_size,
                              hipStream_t stream) {
    const float* feat = (const float*)d_in[0];
    const float* rois = (const float*)d_in[1];
    float* out = (float*)d_out;
    const int K = in_sizes[1] / 5;                 // 512 rois
    dim3 grid(K, CCH / CH_CHUNK, 1);               // 512 x 4 blocks
    roi_align_wmma<<<grid, 256, LDS_BYTES, stream>>>(feat, rois, out, K);
}